// VQExpert_49864570306940
// MI455X (gfx1250) — compile-verified
//
#include <hip/hip_runtime.h>

typedef __attribute__((ext_vector_type(2))) float v2f;
typedef __attribute__((ext_vector_type(8))) float v8f;

#define NUM_CODES   8192
#define CODE_DIM    512
#define BN          32768
#define ROWS_PER_WG 64            // 4 M-tiles of 16
#define N_WGS       512           // BN / ROWS_PER_WG
#define LDS_STRIDE  516           // 512 + 4 pad -> conflict-free b64 LDS reads
#define TILE_LDS    (16 * LDS_STRIDE)
#define ZQ_ELEMS    16777216      // 16*2048*512
#define IDX_ELEMS   32768

// dynamic LDS layout (floats):
#define SM_ZQ     0
#define SM_RED_D  33024
#define SM_RED_I  33536
#define SM_FIDX   34048
#define SM_RED_S  34112
#define SM_FLOATS 34368           // 137472 bytes

// ---------------- kernel 1: codebook squared norms ----------------
__global__ __launch_bounds__(256) void vq_esq_kernel(const float* __restrict__ cb,
                                                     float* __restrict__ esq) {
    __shared__ float buf[256];
    const int row = blockIdx.x;
    const float* r = cb + ((size_t)row << 9);
    float s = 0.f;
    for (int c = threadIdx.x; c < CODE_DIM; c += 256) {
        float v = r[c];
        s = fmaf(v, v, s);
    }
    buf[threadIdx.x] = s;
    __syncthreads();
    for (int st = 128; st > 0; st >>= 1) {
        if (threadIdx.x < st) buf[threadIdx.x] += buf[threadIdx.x + st];
        __syncthreads();
    }
    if (threadIdx.x == 0) esq[row] = buf[0];
}

// ---------------- kernel 2: fused GEMM + argmin + gather + loss ----------------
__global__ __launch_bounds__(256) void vq_main_kernel(const float* __restrict__ z_e,
                                                      const float* __restrict__ cb,
                                                      const float* __restrict__ esq,
                                                      float* __restrict__ zq_out,
                                                      int* __restrict__ idx_out,
                                                      float* __restrict__ partials) {
    extern __shared__ float smem[];
    float* sm_z      = smem + SM_ZQ;
    float* red_d     = smem + SM_RED_D;
    int*   red_i     = (int*)(smem + SM_RED_I);
    int*   final_idx = (int*)(smem + SM_FIDX);
    float* red_s     = smem + SM_RED_S;

    const int tid = threadIdx.x;
    const int mt  = blockIdx.x;

    // ---- stage 64x512 z block into LDS (padded rows) ----
    const float* zbase = z_e + ((size_t)mt << 15);       // mt * 64 * 512
    for (int i = tid; i < (ROWS_PER_WG * CODE_DIM) / 4; i += 256) {
        float4 v = ((const float4*)zbase)[i];
        int e = i << 2;
        int row = e >> 9;                                // 0..63
        int col = e & 511;
        float* d = &sm_z[row * LDS_STRIDE + col];
        d[0] = v.x; d[1] = v.y; d[2] = v.z; d[3] = v.w;
    }
    __syncthreads();

    const int wave  = tid >> 5;
    const int lane  = tid & 31;
    const int lrow  = lane & 15;
    const int khalf = (lane >> 4) << 1;                  // 0 or 2

    float bestd[4][8];
    int   besti[4][8];
#pragma unroll
    for (int u = 0; u < 4; ++u)
#pragma unroll
        for (int r = 0; r < 8; ++r) { bestd[u][r] = 3.4e38f; besti[u][r] = 0x7fffffff; }

    // loop-invariant A fragment base pointers: one base VGPR per M sub-tile so
    // the unrolled ds loads use small immediate offsets (and 2addr pairing)
    const int abase = lrow * LDS_STRIDE + khalf;
    const float* __restrict__ ap0 = sm_z + abase;
    const float* __restrict__ ap1 = sm_z + abase + 1 * TILE_LDS;
    const float* __restrict__ ap2 = sm_z + abase + 2 * TILE_LDS;
    const float* __restrict__ ap3 = sm_z + abase + 3 * TILE_LDS;

    // each wave owns 64 N-tiles: [wave*64, wave*64+64)
    for (int t = 0; t < 64; ++t) {
        const int n0 = (wave * 64 + t) << 4;             // base code index of tile
        const float* __restrict__ bp = cb + (((size_t)(n0 + lrow)) << 9) + khalf;
        const float esqv = esq[n0 + lrow];

        v8f acc0 = {0.f,0.f,0.f,0.f,0.f,0.f,0.f,0.f};
        v8f acc1 = {0.f,0.f,0.f,0.f,0.f,0.f,0.f,0.f};
        v8f acc2 = {0.f,0.f,0.f,0.f,0.f,0.f,0.f,0.f};
        v8f acc3 = {0.f,0.f,0.f,0.f,0.f,0.f,0.f,0.f};

#pragma unroll 8
        for (int k = 0; k < CODE_DIM; k += 4) {
            v2f b  = *(const v2f*)(bp  + k);
            v2f a0 = *(const v2f*)(ap0 + k);
            v2f a1 = *(const v2f*)(ap1 + k);
            v2f a2 = *(const v2f*)(ap2 + k);
            v2f a3 = *(const v2f*)(ap3 + k);
            acc0 = __builtin_amdgcn_wmma_f32_16x16x4_f32(false, a0, false, b, (short)0, acc0, false, false);
            acc1 = __builtin_amdgcn_wmma_f32_16x16x4_f32(false, a1, false, b, (short)0, acc1, false, false);
            acc2 = __builtin_amdgcn_wmma_f32_16x16x4_f32(false, a2, false, b, (short)0, acc2, false, false);
            acc3 = __builtin_amdgcn_wmma_f32_16x16x4_f32(false, a3, false, b, (short)0, acc3, false, false);
        }

        const int n = n0 + lrow;
#pragma unroll
        for (int r = 0; r < 8; ++r) {
            float d0 = fmaf(-2.f, acc0[r], esqv);
            if (d0 < bestd[0][r]) { bestd[0][r] = d0; besti[0][r] = n; }
            float d1 = fmaf(-2.f, acc1[r], esqv);
            if (d1 < bestd[1][r]) { bestd[1][r] = d1; besti[1][r] = n; }
            float d2 = fmaf(-2.f, acc2[r], esqv);
            if (d2 < bestd[2][r]) { bestd[2][r] = d2; besti[2][r] = n; }
            float d3 = fmaf(-2.f, acc3[r], esqv);
            if (d3 < bestd[3][r]) { bestd[3][r] = d3; besti[3][r] = n; }
        }
    }

    // ---- reduce argmin across the 16 lanes of each half-wave (N groups) ----
#pragma unroll
    for (int u = 0; u < 4; ++u) {
#pragma unroll
        for (int r = 0; r < 8; ++r) {
            float d_ = bestd[u][r];
            int   i_ = besti[u][r];
#pragma unroll
            for (int off = 8; off > 0; off >>= 1) {
                float od = __shfl_xor(d_, off, 16);
                int   oi = __shfl_xor(i_, off, 16);
                if (od < d_ || (od == d_ && oi < i_)) { d_ = od; i_ = oi; }
            }
            bestd[u][r] = d_;
            besti[u][r] = i_;
        }
    }
    if (lrow == 0) {
        const int half = lane >> 4;                      // rows r (half 0) / 8+r (half 1)
#pragma unroll
        for (int u = 0; u < 4; ++u)
#pragma unroll
            for (int r = 0; r < 8; ++r) {
                int row = u * 16 + half * 8 + r;         // 0..63 within WG
                red_d[wave * 64 + row] = bestd[u][r];
                red_i[wave * 64 + row] = besti[u][r];
            }
    }
    __syncthreads();

    // ---- reduce across waves; emit indices ----
    if (tid < ROWS_PER_WG) {
        float bd = red_d[tid];
        int   bi = red_i[tid];
        for (int w = 1; w < 8; ++w) {
            float d  = red_d[w * 64 + tid];
            int   i2 = red_i[w * 64 + tid];
            if (d < bd || (d == bd && i2 < bi)) { bd = d; bi = i2; }
        }
        final_idx[tid] = bi;
        idx_out[mt * ROWS_PER_WG + tid] = bi;
    }
    __syncthreads();

    // ---- gather z_q, write straight-through output, accumulate loss ----
    float lsum = 0.f;
    float* zq_base = zq_out + ((size_t)mt << 15);
    for (int e = tid; e < ROWS_PER_WG * CODE_DIM; e += 256) {
        int row = e >> 9;
        int col = e & 511;
        float zq = cb[((size_t)final_idx[row] << 9) + col];
        float ze = sm_z[row * LDS_STRIDE + col];
        float diff = zq - ze;                            // match reference rounding:
        zq_base[e] = ze + diff;                          // z_e + (z_q - z_e)
        lsum = fmaf(diff, diff, lsum);
    }
    red_s[tid] = lsum;
    __syncthreads();
    for (int st = 128; st > 0; st >>= 1) {
        if (tid < st) red_s[tid] += red_s[tid + st];
        __syncthreads();
    }
    if (tid == 0) partials[mt] = red_s[0];               // deterministic: own slot, no atomics
}

// ---------------- kernel 3: loss finalize ----------------
__global__ __launch_bounds__(256) void vq_finalize_kernel(const float* __restrict__ partials,
                                                          float* __restrict__ loss_out) {
    __shared__ float buf[256];
    float s = 0.f;
    for (int i = threadIdx.x; i < N_WGS; i += 256) s += partials[i];
    buf[threadIdx.x] = s;
    __syncthreads();
    for (int st = 128; st > 0; st >>= 1) {
        if (threadIdx.x < st) buf[threadIdx.x] += buf[threadIdx.x + st];
        __syncthreads();
    }
    if (threadIdx.x == 0) {
        // (codebook_loss + 0.25*commit_loss)/C = S/16777216 * 1.25 / 512
        double v = (double)buf[0] * (1.25 / (512.0 * 16777216.0));
        loss_out[0] = (float)v;
    }
}

extern "C" void kernel_launch(void* const* d_in, const int* in_sizes, int n_in,
                              void* d_out, int out_size, void* d_ws, size_t ws_size,
                              hipStream_t stream) {
    const float* z_e = (const float*)d_in[0];
    const float* cb  = (const float*)d_in[1];

    float* out      = (float*)d_out;
    float* zq_out   = out;
    int*   idx_out  = (int*)(out + ZQ_ELEMS);
    float* loss_out = out + ZQ_ELEMS + IDX_ELEMS;

    float* esq      = (float*)d_ws;          // 8192 floats
    float* partials = esq + NUM_CODES;       // 512 floats

    vq_esq_kernel<<<NUM_CODES, 256, 0, stream>>>(cb, esq);
    vq_main_kernel<<<N_WGS, 256, SM_FLOATS * sizeof(float), stream>>>(z_e, cb, esq,
                                                                     zq_out, idx_out, partials);
    vq_finalize_kernel<<<1, 256, 0, stream>>>(partials, loss_out);
}